// kernel_generated_1_62904091017324
// MI455X (gfx1250) — compile-verified
//
#include <hip/hip_runtime.h>

// CDNA5 (gfx1250) fp32 WMMA implementation of the rolled grouped 1-D conv.
//
//   y[img,i,n,m] = sum_j sum_k W[j,k,i] * xpad[img,j, n (h-roll at store), m+k (w-roll in staging)]
//
// - Staging: one TENSOR_LOAD_TO_LDS (TDM) per workgroup pulls the 2x128x28
//   fp32 row slab into LDS (raw), then an LDS->LDS pass builds the
//   width-rolled zero-padded B layout.
// - Compute: V_WMMA_F32_16X16X4_F32, A = W^T tile (out-chan x in-chan),
//   B = padded row (in-chan x width). Each wave owns TWO 16-chan M tiles so
//   every B fragment feeds two WMMAs (1 LDS load per WMMA).

typedef __attribute__((ext_vector_type(2))) float        v2f;
typedef __attribute__((ext_vector_type(8))) float        v8f;
typedef __attribute__((ext_vector_type(4))) unsigned int v4u;
typedef __attribute__((ext_vector_type(8))) int          v8i;
typedef __attribute__((ext_vector_type(4))) int          v4i;

#define N_IMG   256   // B*G
#define CG      128   // channels per group (in and out)
#define H       28
#define WD      28
#define RPB     2     // conv rows per workgroup
#define USTRIDE 32    // padded LDS row stride (floats)
#define TAPS    3

__global__ __launch_bounds__(128)
void shiftconv_wmma_f32(const float* __restrict__ x,
                        const float* __restrict__ Wt,   // (j=128, k=3, i=128)
                        float* __restrict__ out)
{
    __shared__ float raw[RPB * CG * WD];                 // TDM destination: [j][n][m], 28 KB
    __shared__ float xs[RPB * CG * USTRIDE + 4];         // padded rolled rows: [r][j][u], 32 KB

    const int img = blockIdx.x / (H / RPB);
    const int rb  = blockIdx.x % (H / RPB);
    const int n0  = rb * RPB;

    const int tid  = threadIdx.x;
    const int lane = tid & 31;
    const int wave = tid >> 5;

    __builtin_prefetch(Wt, 0, 3);                        // keep W hot (global_prefetch_b8)

    // ---- TDM: bulk-load the row slab into LDS (wave 0 issues, EXEC-independent) ----
    if (tid == 0) {
        const unsigned lds_off = (unsigned)(unsigned long long)(const void*)raw;
        const unsigned long long ga =
            (unsigned long long)(const void*)(x + (size_t)img * CG * H * WD + (size_t)n0 * WD);

        // D# group 0: count=1 | lds_addr | global_addr[56:0] | type=2
        v4u g0 = { 1u,
                   lds_off,
                   (unsigned)(ga & 0xFFFFFFFFu),
                   (unsigned)((ga >> 32) & 0x01FFFFFFu) | (2u << 30) };
        // D# group 1: data_size=4B; dims (m=28, n=2, j=128); tile (28,2,128);
        //             dim0_stride=28 (n step), dim1_stride=784 (j step)
        v8i g1 = { (int)(2u << 16),                      // wg_mask=0, data_size=2 (4 bytes)
                   (int)(WD << 16),                      // tensor_dim0[15:0] @ bits63:48
                   (int)(RPB << 16),                     // tensor_dim1[15:0] @ bits95:80
                   (int)(WD << 16),                      // tile_dim0=28 @ bits127:112
                   (int)(RPB | (CG << 16)),              // tile_dim1=2, tile_dim2=128
                   (int)WD,                              // tensor_dim0_stride[31:0]=28
                   (int)((WD * H) << 16),                // dim0_stride hi=0 | dim1_stride[15:0]=784
                   0 };                                  // dim1_stride[47:16]=0
        // D# group 2: tensor_dim2=128, no dim3, iterate off
        v4i g2 = { CG, 0, 0, 0 };
        // D# group 3: unused (zero)
        v4i g3 = { 0, 0, 0, 0 };
        // Trailing descriptor words (6-arg clang-23 form): zero-filled
        v8i g4 = { 0, 0, 0, 0, 0, 0, 0, 0 };

        __builtin_amdgcn_tensor_load_to_lds(g0, g1, g2, g3, g4, 0);
        __builtin_amdgcn_s_wait_tensorcnt(0);
    }
    __syncthreads();

    // Make the TDM-written LDS visible to the optimizer (it never sees a store to raw).
    const float* rawp = raw;
    asm volatile("" : "+v"(rawp) : : "memory");

    // ---- Build width-rolled + zero-padded rows:  xs[(r*CG+j)*32 + u] ----
    //   u==0 -> 0 ; u in [1,29): x[.., (u-2+28)%28] ; u>=29 -> 0
    for (int slot = tid; slot < RPB * CG * USTRIDE; slot += 128) {
        const int u = slot & (USTRIDE - 1);
        const int j = (slot >> 5) & (CG - 1);
        const int r = slot >> 12;
        float v = 0.0f;
        if (u >= 1 && u < 29) {
            const int m = (u - 2 + WD) % WD;
            v = rawp[(j * RPB + r) * WD + m];
        }
        xs[slot] = v;
    }
    __syncthreads();

    // ---- WMMA main loop ----
    // A (16x4): lanes 0-15: M=lane, v0=K0,v1=K1 ; lanes 16-31: M=lane-16, v0=K2,v1=K3
    // B (4x16): v0: lanes 0-15 row K0 / lanes 16-31 row K1 ; v1: rows K2/K3
    const int i_wave = wave * 32;                        // 2 M tiles per wave
    const int nl = lane & 15;
    const int hp = lane >> 4;

    v8f acc[2][RPB][2];                                  // [mt][r][nt]
    #pragma unroll
    for (int mt = 0; mt < 2; ++mt)
        #pragma unroll
        for (int r = 0; r < RPB; ++r)
            #pragma unroll
            for (int nt = 0; nt < 2; ++nt)
                acc[mt][r][nt] = (v8f){};

    for (int k = 0; k < TAPS; ++k) {
        for (int jc4 = 0; jc4 < CG / 4; ++jc4) {
            const int jc = jc4 * 4;

            // Two A fragments (W slices, L2-resident).
            const int j0 = jc + hp * 2;
            const float* wp = Wt + (size_t)j0 * (TAPS * CG) + k * CG + i_wave + nl;
            v2f a[2];
            #pragma unroll
            for (int mt = 0; mt < 2; ++mt) {
                a[mt].x = wp[mt * 16];
                a[mt].y = wp[mt * 16 + TAPS * CG];
            }

            #pragma unroll
            for (int r = 0; r < RPB; ++r) {
                #pragma unroll
                for (int nt = 0; nt < 2; ++nt) {
                    const int u = nt * 16 + nl + k;
                    const float* bp = &xs[((r * CG) + jc + hp) * USTRIDE + u];
                    v2f b;
                    b.x = bp[0];
                    b.y = bp[2 * USTRIDE];
                    #pragma unroll
                    for (int mt = 0; mt < 2; ++mt) {
                        acc[mt][r][nt] = __builtin_amdgcn_wmma_f32_16x16x4_f32(
                            false, a[mt], false, b, (short)0, acc[mt][r][nt], false, false);
                    }
                }
            }
        }
    }

    // ---- Store with the height roll applied (n_out = n_conv + 1 mod 28) ----
    // C/D layout: VGPR v: lanes 0-15 -> (M=v, N=lane) ; lanes 16-31 -> (M=v+8, N=lane-16)
    #pragma unroll
    for (int mt = 0; mt < 2; ++mt) {
        #pragma unroll
        for (int r = 0; r < RPB; ++r) {
            const int n_out = (n0 + r + 1) % H;
            #pragma unroll
            for (int nt = 0; nt < 2; ++nt) {
                const int m = nt * 16 + nl;
                if (m < WD) {
                    #pragma unroll
                    for (int v = 0; v < 8; ++v) {
                        const int i = i_wave + mt * 16 + v + hp * 8;
                        out[(((size_t)img * CG + i) * H + n_out) * WD + m] = acc[mt][r][nt][v];
                    }
                }
            }
        }
    }
}

extern "C" void kernel_launch(void* const* d_in, const int* in_sizes, int n_in,
                              void* d_out, int out_size, void* d_ws, size_t ws_size,
                              hipStream_t stream) {
    (void)in_sizes; (void)n_in; (void)out_size; (void)d_ws; (void)ws_size;
    const float* x = (const float*)d_in[0];
    const float* W = (const float*)d_in[1];
    float* out = (float*)d_out;

    dim3 grid(N_IMG * (H / RPB));   // 256 images * 14 row-blocks = 3584 workgroups
    dim3 block(128);                // 4 wave32 waves, 2 M-tiles each -> 128 out channels
    hipLaunchKernelGGL(shiftconv_wmma_f32, grid, block, 0, stream, x, W, out);
}